// Net_rec_10058813407895
// MI455X (gfx1250) — compile-verified
//
#include <hip/hip_runtime.h>
#include <hip/hip_bf16.h>
#include <stdint.h>

#define B_    64
#define D_    256
#define HW_   25
#define NPAIR 32896   // 256*257/2
#define JC    4       // j-quarter blocks for rowsum kernel
#define PC    16      // pair-chunk blocks for output kernel

typedef __attribute__((ext_vector_type(2))) float v2f;
typedef __attribute__((ext_vector_type(8))) float v8f;

// Stage a contiguous f32 block from global memory into LDS using the CDNA5
// async global->LDS path (ASYNCcnt-tracked). GVS addressing:
//   global addr = SADDR(64b) + VADDR(32b offset);  VDST = LDS byte address.
__device__ __forceinline__ void stage_f_async(const float* __restrict__ gbase,
                                              float* lds, int nElems,
                                              int tid, int nthreads) {
  unsigned ldsBase = (unsigned)(uintptr_t)lds;   // low 32 bits = LDS offset
  for (int e = tid; e < nElems; e += nthreads) {
    unsigned loff = ldsBase + 4u * (unsigned)e;
    unsigned goff = 4u * (unsigned)e;
    asm volatile("global_load_async_to_lds_b32 %0, %1, %2"
                 :: "v"(loff), "v"(goff), "s"(gbase) : "memory");
  }
  asm volatile("s_wait_asynccnt 0" ::: "memory");
}

// Kernel 1: per-batch row sums of the (pre-scale) relation matrix.
// Wave w handles rows [16w, 16w+16). Block (b, q) handles j in [64q, 64q+64).
// Each lane accumulates 2 of the 4 K-slot partials; one V_WMMA_F32_16X16X4_F32
// with A = ones(16x4) reduces all 4 slots across both half-waves:
// D[m,n] = sum_k B[k,n]  ->  full 64-j partial rowsum per column n.
__global__ void __launch_bounds__(512) rowsum_kernel(const float* __restrict__ feat,
                                                     float* __restrict__ mpart) {
  __shared__ float fL[D_ * HW_];
  const int b = blockIdx.x, q = blockIdx.y;
  const int tid = threadIdx.x;

  stage_f_async(feat + (size_t)b * D_ * HW_, fL, D_ * HW_, tid, 512);
  __syncthreads();

  const int wave = tid >> 5, lane = tid & 31;
  const int i = (wave << 4) + (lane & 15);

  float fr[HW_];
#pragma unroll
  for (int p = 0; p < HW_; ++p) fr[p] = fL[i * HW_ + p];

  // lanes 0-15: j slots [0,16) and [16,32); lanes 16-31: [32,48) and [48,64)
  const int jbase = q * 64 + ((lane & 16) << 1);
  float acc0 = 0.f, acc1 = 0.f;
  for (int jj = 0; jj < 16; ++jj) {
    const float* fj0 = &fL[(jbase + jj) * HW_];
    const float* fj1 = &fL[(jbase + 16 + jj) * HW_];
#pragma unroll
    for (int p = 0; p < HW_; ++p) {
      float a = fr[p];
      float b0 = fj0[p];
      float b1 = fj1[p];
      acc0 += fabsf(a + b0) - fabsf(a - b0);   // |a+b| - |a-b|
      acc1 += fabsf(a + b1) - fabsf(a - b1);
    }
  }

  v2f av; av.x = 1.0f; av.y = 1.0f;           // A = ones(16x4)
  v2f bv; bv.x = acc0; bv.y = acc1;           // B = 4x16 slot partials
  v8f cv = {};
  v8f dv = __builtin_amdgcn_wmma_f32_16x16x4_f32(
      /*neg_a=*/false, av, /*neg_b=*/false, bv,
      /*c_mod=*/(short)0, cv, /*reuse_a=*/false, /*reuse_b=*/false);

  // D VGPR0, lanes 0-15 hold (M=0, N=lane) = rowsum for row 16*wave + lane.
  if (lane < 16) {
    mpart[((size_t)b * JC + q) * D_ + (wave << 4) + lane] = dv[0];
  }
}

// Kernel 2: per-pair output. centered = (g_ij - (m_i + m_j)/256) * 0.5*exp(T)
__global__ void __launch_bounds__(256) out_kernel(const float* __restrict__ feat,
                                                  const float* __restrict__ temp,
                                                  const float* __restrict__ mpart,
                                                  float* __restrict__ out) {
  __shared__ float fL[D_ * HW_];
  __shared__ float mL[D_];
  const int b = blockIdx.x, pcId = blockIdx.y;
  const int tid = threadIdx.x;

  stage_f_async(feat + (size_t)b * D_ * HW_, fL, D_ * HW_, tid, 256);
  {
    const float* mp = mpart + (size_t)b * JC * D_;
    mL[tid] = mp[tid] + mp[D_ + tid] + mp[2 * D_ + tid] + mp[3 * D_ + tid];
  }
  __syncthreads();

  const float scale = 0.5f * __expf(temp[0]);

  for (int k = pcId * 256 + tid; k < NPAIR; k += PC * 256) {
    // invert row-major triu index: S(i) = 256i - i(i-1)/2 <= k < S(i+1)
    float disc = 263169.0f - 8.0f * (float)k;   // 513^2 - 8k (exact in f32)
    int i = (int)((513.0f - sqrtf(disc)) * 0.5f);
    if (i > 255) i = 255;
    if (i < 0) i = 0;
    while (i > 0 && (i * 256 - (i * (i - 1)) / 2) > k) --i;
    while (((i + 1) * 256 - ((i + 1) * i) / 2) <= k) ++i;
    const int Si = i * 256 - (i * (i - 1)) / 2;
    const int j = i + (k - Si);

    const float* fi = &fL[i * HW_];
    const float* fj = &fL[j * HW_];
    float acc = 0.f;
#pragma unroll
    for (int p = 0; p < HW_; ++p) {
      float a = fi[p], c = fj[p];
      acc += fabsf(a + c) - fabsf(a - c);
    }
    out[(size_t)b * NPAIR + k] =
        (acc - (mL[i] + mL[j]) * (1.0f / 256.0f)) * scale;
  }
}

extern "C" void kernel_launch(void* const* d_in, const int* in_sizes, int n_in,
                              void* d_out, int out_size, void* d_ws, size_t ws_size,
                              hipStream_t stream) {
  const float* feat = (const float*)d_in[0];   // [64, 256, 5, 5] f32
  const float* temp = (const float*)d_in[1];   // [1, 1] f32
  float* mpart = (float*)d_ws;                 // [64][4][256] f32 partial rowsums (256 KB)

  rowsum_kernel<<<dim3(B_, JC), 512, 0, stream>>>(feat, mpart);
  out_kernel<<<dim3(B_, PC), 256, 0, stream>>>(feat, temp, mpart, (float*)d_out);
}